// GCN_3959959847143
// MI455X (gfx1250) — compile-verified
//
#include <hip/hip_runtime.h>
#include <stdint.h>

// -------- CDNA5 (gfx1250) fp32 WMMA fragment types --------
typedef __attribute__((ext_vector_type(2))) float v2f;  // A/B frag of V_WMMA_F32_16X16X4_F32
typedef __attribute__((ext_vector_type(8))) float v8f;  // C/D frag (16x16 fp32)

// =====================================================================
// GEMM: C[MxN] = op(A[MxK]) @ B[KxN], row-major fp32. M%16==0.
// N, K, KP compile-time. One wave32 computes a 16(M) x 64(N) tile with
// four V_WMMA_F32_16X16X4_F32 accumulators sharing one A fragment.
//
// B is staged per 64-row K-panel into LDS in "pair layout":
//   ldsB[(p*N + n)*2 + {0,1}] = { B[k0+2p][n], B[k0+2p+1][n] }
// so lane (h = lane>>4, l = lane&15) fetches its B fragment for column n
// (rows k+2h, k+2h+1) as ONE aligned ds_load_b64, and the 4 N-subtiles
// are immediate DS offsets (+128B apart).
//
// ISA fragment layouts (CDNA5 ISA):
//   A 16x4 : lane l<16 -> row l, {K=0,K=1}; lane 16+l -> row l, {K=2,K=3}
//   C/D    : v[i]: lanes 0-15 -> (M=i, N=lane); lanes 16-31 -> (M=8+i, N=lane-16)
// =====================================================================
template <bool RELU_A, int N, int K, int KP>
__global__ __launch_bounds__(256)
void gcn_gemm_wmma_f32(const float* __restrict__ A,
                       const float* __restrict__ B,
                       float* __restrict__ C,
                       int M)
{
    extern __shared__ float ldsB[];            // KP * N floats (pair layout)

    const int tid  = threadIdx.x;
    const int lane = tid & 31;
    const int wave = tid >> 5;                 // 8 waves / block

    constexpr int NCOLS = N / 64;              // 16x64 wave tiles per M-row
    const int gwave = blockIdx.x * 8 + wave;
    const int mt    = gwave / NCOLS;
    const int n0    = (gwave - mt * NCOLS) * 64;
    const bool active = (mt * 16 < M);
    const int m0    = mt * 16;

    const int half = lane >> 4;                // 0: K rows 0..1, 1: K rows 2..3
    const int l    = lane & 15;

    const float* arow  = active ? (A + (size_t)(m0 + l) * K + 2 * half) : A;
    const float* bfrag = ldsB + (size_t)(half * N + n0 + l) * 2;

    v8f acc0 = {}, acc1 = {}, acc2 = {}, acc3 = {};

    for (int k0 = 0; k0 < K; k0 += KP) {
        if (k0 != 0) __syncthreads();          // panel reuse WAR
        // ---- stage B panel rows [k0, k0+KP) into LDS (pair layout) ----
        constexpr int PAIRS = (KP / 2) * N;    // 64-row panel -> KP/2 * N pairs
        #pragma unroll
        for (int idx = 0; idx < PAIRS / 256; ++idx) {
            const int pi = idx * 256 + tid;    // pair-linear index = p*N + n
            const int p  = pi / N;             // N is a power of two (template)
            const int n  = pi - p * N;
            const int gk = k0 + 2 * p;
            v2f pr;
            pr.x = B[(size_t)gk * N + n];
            pr.y = B[(size_t)(gk + 1) * N + n];
            *(v2f*)(ldsB + (size_t)pi * 2) = pr;   // 8B-stride: conflict-free
        }
        __syncthreads();

        if (active) {
            #pragma unroll
            for (int kk = 0; kk < KP; kk += 4) {
                float a0 = arow[k0 + kk];
                float a1 = arow[k0 + kk + 1];
                if (RELU_A) { a0 = fmaxf(a0, 0.0f); a1 = fmaxf(a1, 0.0f); }
                v2f a; a.x = a0; a.y = a1;

                const float* bp = bfrag + (kk >> 1) * (N * 2);
                v2f b0 = *(const v2f*)(bp);        // cols n0+l
                v2f b1 = *(const v2f*)(bp + 32);   // cols n0+16+l
                v2f b2 = *(const v2f*)(bp + 64);   // cols n0+32+l
                v2f b3 = *(const v2f*)(bp + 96);   // cols n0+48+l

                acc0 = __builtin_amdgcn_wmma_f32_16x16x4_f32(false, a, false, b0,
                                                             (short)0, acc0, false, false);
                acc1 = __builtin_amdgcn_wmma_f32_16x16x4_f32(false, a, false, b1,
                                                             (short)0, acc1, false, false);
                acc2 = __builtin_amdgcn_wmma_f32_16x16x4_f32(false, a, false, b2,
                                                             (short)0, acc2, false, false);
                acc3 = __builtin_amdgcn_wmma_f32_16x16x4_f32(false, a, false, b3,
                                                             (short)0, acc3, false, false);
            }
        }
    }

    if (active) {
        #pragma unroll
        for (int i = 0; i < 8; ++i) {
            const size_t r = (size_t)(m0 + i + 8 * half);
            float* crow = C + r * N + n0 + l;
            crow[0]  = acc0[i];
            crow[16] = acc1[i];
            crow[32] = acc2[i];
            crow[48] = acc3[i];
        }
    }
}

// =====================================================================
// out[n][d] = bias[d], vectorized b128. D (=dmask+1) is 128 or 64.
// =====================================================================
__global__ __launch_bounds__(256)
void gcn_bias_init(float* __restrict__ out, const float* __restrict__ bias,
                   size_t total4, int dmask)
{
    size_t i = (size_t)blockIdx.x * blockDim.x + threadIdx.x;
    const size_t stride = (size_t)gridDim.x * blockDim.x;
    for (; i < total4; i += stride) {
        const int d = (int)((i * 4) & (size_t)dmask);
        *(float4*)(out + i * 4) = *(const float4*)(bias + d);
    }
}

// =====================================================================
// COO SPMM scatter: out[row] += w * sup[col]
// One wave32 per edge; lane owns VEC consecutive features -> the gather
// is one global_load_b128/b64 per lane (512B/256B coalesced per wave);
// scatter via non-returning global_atomic_add_f32.
// =====================================================================
template <int VEC>
__global__ __launch_bounds__(256)
void gcn_spmm_scatter(const int* __restrict__ rows,
                      const int* __restrict__ cols,
                      const float* __restrict__ w,
                      const float* __restrict__ sup,
                      float* __restrict__ out,
                      int E, int D)
{
    typedef __attribute__((ext_vector_type(VEC))) float vec_t;
    const int lane = threadIdx.x & 31;
    int e = blockIdx.x * 8 + (threadIdx.x >> 5);
    const int estride = gridDim.x * 8;

    for (; e < E; e += estride) {
        const int   r  = rows[e];
        const int   c  = cols[e];
        const float wt = w[e];

        vec_t s = *(const vec_t*)(sup + (size_t)c * D + lane * VEC);
        float* __restrict__ dst = out + (size_t)r * D + lane * VEC;

        #pragma unroll
        for (int j = 0; j < VEC; ++j)
            atomicAdd(dst + j, wt * s[j]);
    }
}

// =====================================================================
// Launch: GEMM1 -> bias/spmm1 -> GEMM2(ReLU fused) -> bias/spmm2
// =====================================================================
extern "C" void kernel_launch(void* const* d_in, const int* in_sizes, int n_in,
                              void* d_out, int out_size, void* d_ws, size_t ws_size,
                              hipStream_t stream)
{
    const float* x  = (const float*)d_in[0];
    const int*   ei = (const int*)  d_in[1];   // [2, E] flat: rows then cols
    const float* ew = (const float*)d_in[2];
    const float* W1 = (const float*)d_in[3];
    const float* b1 = (const float*)d_in[4];
    const float* W2 = (const float*)d_in[5];
    const float* b2 = (const float*)d_in[6];
    float* out = (float*)d_out;

    const int N  = 100000;              // nodes (== 6250 * 16, no M tail)
    const int E  = in_sizes[2];         // edge count (edge_weight size)
    constexpr int F  = 256;
    constexpr int H1 = 128;
    constexpr int H2 = 64;
    constexpr int KP = 64;              // K-panel rows staged in LDS

    // Workspace layout (fp32): sup1[N*H1] | h[N*H1] | sup2[N*H2]  ~= 128 MB
    float* sup1 = (float*)d_ws;
    float* h    = sup1 + (size_t)N * H1;
    float* sup2 = h    + (size_t)N * H1;

    // ---- Layer 1: sup1 = x @ W1 ----
    {
        const int waves  = (N / 16) * (H1 / 64);       // 12500
        const int blocks = (waves + 7) / 8;            // 1563
        const size_t lds = (size_t)KP * H1 * sizeof(float);   // 32 KB
        gcn_gemm_wmma_f32<false, H1, F, KP><<<blocks, 256, lds, stream>>>(x, W1, sup1, N);
    }
    // h = b1; h[row] += w * sup1[col]
    gcn_bias_init<<<2048, 256, 0, stream>>>(h, b1, (size_t)N * H1 / 4, H1 - 1);
    gcn_spmm_scatter<4><<<8192, 256, 0, stream>>>(ei, ei + E, ew, sup1, h, E, H1);

    // ---- Layer 2: sup2 = relu(h) @ W2 ----
    {
        const int waves  = (N / 16) * (H2 / 64);       // 6250
        const int blocks = (waves + 7) / 8;            // 782
        const size_t lds = (size_t)KP * H2 * sizeof(float);   // 16 KB
        gcn_gemm_wmma_f32<true, H2, H1, KP><<<blocks, 256, lds, stream>>>(h, W2, sup2, N);
    }
    // out = b2; out[row] += w * sup2[col]
    gcn_bias_init<<<2048, 256, 0, stream>>>(out, b2, (size_t)N * H2 / 4, H2 - 1);
    gcn_spmm_scatter<2><<<8192, 256, 0, stream>>>(ei, ei + E, ew, sup2, out, E, H2);
}